// CBDA_8607114461380
// MI455X (gfx1250) — compile-verified
//
#include <hip/hip_runtime.h>

typedef float v2f __attribute__((ext_vector_type(2)));
typedef float v4f __attribute__((ext_vector_type(4)));
typedef float v8f __attribute__((ext_vector_type(8)));

#define B_    64
#define S_    1024
#define D_    128
#define IN_   1024
#define TWO_D 256
#define BLOCKS_ 8

// ---------------------------------------------------------------------------
// Stage 1: gb[64,256] = x_cond[64,1024] @ W[1024,256] + bias, via f32 WMMA.
// One wave per 16x16 output tile; 4 M-tiles x 16 N-tiles = 64 tiles.
// A-frag (16x4 f32): lane{0-15}: v0=K0,v1=K1 ; lane{16-31}: v0=K2,v1=K3 (row=lane%16)
// B-frag (4x16 f32): lane{0-15}: v0=K0,v1=K1 rows ; lane{16-31}: K2,K3 (col=lane%16)
// C/D (16x16 f32):   VGPR r -> M = r + 8*(lane/16), N = lane%16
// ---------------------------------------------------------------------------
__global__ __launch_bounds__(32) void film_gemm_wmma(
    const float* __restrict__ xc,    // [64,1024]
    const float* __restrict__ W,     // [1024,256]
    const float* __restrict__ bias,  // [256]
    float* __restrict__ gb)          // [64,256]
{
    const int tile = blockIdx.x;          // 0..63
    const int m0   = (tile >> 4) << 4;    // 0,16,32,48
    const int n0   = (tile & 15) << 4;    // 0..240
    const int lane = threadIdx.x;         // 0..31 (wave32)
    const int half = lane >> 4;           // 0 or 1
    const int l    = lane & 15;

    // lane-invariant strides; per-lane bases
    const float* aptr = xc + (size_t)(m0 + l) * IN_ + 2 * half;         // A[m][kk+2h..]
    const float* bptr = W  + (size_t)(2 * half) * TWO_D + (n0 + l);     // B[kk+2h][n]

    v8f c = {};
#pragma unroll 4
    for (int kk = 0; kk < IN_; kk += 4) {
        v2f a, bb;
        a.x  = aptr[kk + 0];                        // A[m][kk+2h]
        a.y  = aptr[kk + 1];                        // A[m][kk+2h+1]
        bb.x = bptr[(size_t)kk * TWO_D];            // B[kk+2h][n]
        bb.y = bptr[(size_t)kk * TWO_D + TWO_D];    // B[kk+2h+1][n]
        // 8 args: (neg_a, A, neg_b, B, c_mod, C, reuse_a, reuse_b)
        c = __builtin_amdgcn_wmma_f32_16x16x4_f32(false, a, false, bb,
                                                  (short)0, c, false, false);
    }

    const float bv = bias[n0 + l];
#pragma unroll
    for (int r = 0; r < 8; ++r) {
        const int m = m0 + r + 8 * half;
        gb[(size_t)m * TWO_D + n0 + l] = c[r] + bv;
    }
}

// ---------------------------------------------------------------------------
// Stage 2: out[64,1024,1024], block-diagonal FiLM scatter.
// One 256-thread block per output row (b,s): 1024 floats = one v4f per thread.
// Wave k (k = s>>7) of the 8 waves covers the diagonal block [k*128,(k+1)*128);
// all other waves store zeros. Stores are non-temporal B128: 256 MB streamed
// once, never re-read -> keep it out of L2.
// ---------------------------------------------------------------------------
__global__ __launch_bounds__(256) void film_blockdiag(
    const float* __restrict__ xf,   // [64,1024,128]
    const float* __restrict__ gb,   // [64,256]  (gamma | beta)
    float* __restrict__ out)        // [64,1024,1024]
{
    const int row  = blockIdx.x;        // b*1024 + s
    const int b    = row >> 10;
    const int s    = row & (S_ - 1);
    const int k    = s >> 7;            // diagonal block index
    const int col0 = threadIdx.x << 2;  // 0..1020

    v4f v = {0.f, 0.f, 0.f, 0.f};
    const int d = col0 - (k << 7);      // offset within diagonal block
    if ((unsigned)d < (unsigned)D_) {   // wave-uniform: exactly 1 of 8 waves
        const v4f x  = *(const v4f*)(xf + ((size_t)row << 7) + d);
        const v4f g  = *(const v4f*)(gb + (size_t)b * TWO_D + d);
        const v4f be = *(const v4f*)(gb + (size_t)b * TWO_D + D_ + d);
        v.x = fmaf(1.f + g.x, x.x, be.x);
        v.y = fmaf(1.f + g.y, x.y, be.y);
        v.z = fmaf(1.f + g.z, x.z, be.z);
        v.w = fmaf(1.f + g.w, x.w, be.w);
    }
    v4f* dst = (v4f*)(out + ((size_t)row << 10) + col0);
    __builtin_nontemporal_store(v, dst);
}

// ---------------------------------------------------------------------------
extern "C" void kernel_launch(void* const* d_in, const int* in_sizes, int n_in,
                              void* d_out, int out_size, void* d_ws, size_t ws_size,
                              hipStream_t stream) {
    (void)in_sizes; (void)n_in; (void)out_size; (void)ws_size;

    const float* x_cond = (const float*)d_in[0];  // [64,1024]
    const float* x_film = (const float*)d_in[1];  // [64,1024,128]
    const float* W      = (const float*)d_in[2];  // [1024,256]
    const float* bias   = (const float*)d_in[3];  // [256]
    float* out = (float*)d_out;                   // [64,1024,1024]
    float* gb  = (float*)d_ws;                    // [64,256] scratch (64 KB)

    film_gemm_wmma<<<dim3(64), dim3(32), 0, stream>>>(x_cond, W, bias, gb);
    film_blockdiag<<<dim3(B_ * S_), dim3(256), 0, stream>>>(x_film, gb, out);
}